// GraphConvolutionWithNodeAttrs_6098853560480
// MI455X (gfx1250) — compile-verified
//
#include <hip/hip_runtime.h>
#include <hip/hip_bf16.h>

typedef __attribute__((ext_vector_type(2))) float v2f;
typedef __attribute__((ext_vector_type(8))) float v8f;

#define NN 20000
#define EE 320000

// ---------------------------------------------------------------------------
// Kernel 0: pre-scale + transpose FC weights into column-major pair layout.
//   W1cm[n*16 + k] = Wfc1[k][n] / sqrt(16)   (n<64, k<16)
//   W2cm[n*64 + k] = Wfc2[k][n] / sqrt(64)   (n<192, k<64)
// ---------------------------------------------------------------------------
__global__ void wprep_kernel(const float* __restrict__ Wfc1,
                             const float* __restrict__ Wfc2,
                             float* __restrict__ W1cm,
                             float* __restrict__ W2cm) {
    int i = blockIdx.x * blockDim.x + threadIdx.x;
    if (i < 64 * 16) {
        int n = i >> 4, k = i & 15;
        W1cm[i] = Wfc1[k * 64 + n] * 0.25f;
    }
    if (i < 192 * 64) {
        int n = i >> 6, k = i & 63;
        W2cm[i] = Wfc2[k * 192 + n] * 0.125f;
    }
}

// ---------------------------------------------------------------------------
// Kernel 1: per-node linears.
//   f[n][0:64]          = (x0 @ Wl1_0) * a / sqrt(64)
//   f[n][64 + u*3 + m]  = (sum_u0 x1[n][u0][m] * Wl1_1[u0][u]) * a / sqrt(32)
//   sc: same with Wsc0 / Wsc1.
// ---------------------------------------------------------------------------
__global__ __launch_bounds__(160) void node_kernel(
    const float* __restrict__ x, const float* __restrict__ na,
    const float* __restrict__ Wsc0, const float* __restrict__ Wsc1,
    const float* __restrict__ Wl1_0, const float* __restrict__ Wl1_1,
    float* __restrict__ f, float* __restrict__ sc) {
    __shared__ float xr[160];
    const int n = blockIdx.x;
    const int tid = threadIdx.x;
    xr[tid] = x[(size_t)n * 160 + tid];
    __syncthreads();
    const float a = na[n];
    const float inv8 = 0.125f;                       // 1/sqrt(64)
    const float inv32 = 0.17677669529663688110f;     // 1/sqrt(32)
    if (tid < 64) {
        float accf = 0.f, accs = 0.f;
        #pragma unroll 8
        for (int u = 0; u < 64; ++u) {
            float xv = xr[u];
            accf += xv * Wl1_0[u * 64 + tid];
            accs += xv * Wsc0[u * 64 + tid];
        }
        f[(size_t)n * 160 + tid]  = accf * a * inv8;
        sc[(size_t)n * 160 + tid] = accs * a * inv8;
    } else {
        int j = tid - 64, v = j / 3, m = j - v * 3;
        float accf = 0.f, accs = 0.f;
        #pragma unroll 8
        for (int u = 0; u < 32; ++u) {
            float xv = xr[64 + u * 3 + m];
            accf += xv * Wl1_1[u * 32 + v];
            accs += xv * Wsc1[u * 32 + v];
        }
        f[(size_t)n * 160 + 64 + v * 3 + m]  = accf * a * inv32;
        sc[(size_t)n * 160 + 64 + v * 3 + m] = accs * a * inv32;
    }
}

// ---------------------------------------------------------------------------
// Kernel 2: edge MLP via fp32 WMMA + tensor products + atomic scatter.
// 4 waves/block, 32 edges/wave (two 16-row subtiles), 128 edges/block.
// acc layout per node (384 floats): [0:64]=n0, [64:96]=n1, [96:384]=vec(u*3+m)
// ---------------------------------------------------------------------------
__global__ __launch_bounds__(128) void edge_kernel(
    const float* __restrict__ es,    // E x 16
    const float* __restrict__ sh,    // E x 4
    const int* __restrict__ esrc, const int* __restrict__ edst,
    const float* __restrict__ W1cm,  // 64 x 16  (col-major pairs)
    const float* __restrict__ W2cm,  // 192 x 64 (col-major pairs)
    const float* __restrict__ f,     // N x 160
    float* __restrict__ acc,         // N x 384
    float* __restrict__ cnt)         // N
{
    __shared__ float Hlds[4][32 * 64];
    const int wave = threadIdx.x >> 5;
    const int lane = threadIdx.x & 31;
    const int lo = lane & 15;       // col / row-within-16
    const int hi = lane >> 4;       // K-half / M-half select
    const int ebase = blockIdx.x * 128 + wave * 32;
    float* H = Hlds[wave];

    // ---- Phase 1: H = silu(ES @ W1) ----
    for (int s = 0; s < 2; ++s) {
        v2f a[4];
        const float* esrow = es + (size_t)(ebase + s * 16 + lo) * 16;
        #pragma unroll
        for (int ks = 0; ks < 4; ++ks) {
            int k = 4 * ks + 2 * hi;
            a[ks].x = esrow[k];
            a[ks].y = esrow[k + 1];
        }
        #pragma unroll
        for (int t = 0; t < 4; ++t) {
            v8f c = {};
            #pragma unroll
            for (int ks = 0; ks < 4; ++ks) {
                int k = 4 * ks + 2 * hi;
                const float* bp = W1cm + (16 * t + lo) * 16 + k;
                v2f b; b.x = bp[0]; b.y = bp[1];
                c = __builtin_amdgcn_wmma_f32_16x16x4_f32(
                        false, a[ks], false, b, (short)0, c, false, false);
            }
            #pragma unroll
            for (int r = 0; r < 8; ++r) {
                float y = c[r];
                H[(s * 16 + r + 8 * hi) * 64 + 16 * t + lo] =
                    y / (1.0f + __expf(-y));
            }
        }
    }
    __syncthreads();

    // ---- Phase 2: W = H @ W2, then tensor products + scatter ----
    const float inv_sqrt3 = 0.57735026918962576451f;
    for (int s = 0; s < 2; ++s) {
        v2f a[16];
        const float* hrow = H + (s * 16 + lo) * 64;
        #pragma unroll
        for (int ks = 0; ks < 16; ++ks) {
            int k = 4 * ks + 2 * hi;
            a[ks].x = hrow[k];
            a[ks].y = hrow[k + 1];
        }
        int srcs[8], dsts[8];
        float s0[8], s1x[8], s1y[8], s1z[8];
        #pragma unroll
        for (int r = 0; r < 8; ++r) {
            int e = ebase + s * 16 + r + 8 * hi;
            srcs[r] = esrc[e];
            dsts[r] = edst[e];
            const float* shp = sh + (size_t)e * 4;
            s0[r] = shp[0]; s1x[r] = shp[1]; s1y[r] = shp[2]; s1z[r] = shp[3];
        }
        for (int t = 0; t < 12; ++t) {   // uniform tile loop
            v8f c = {};
            #pragma unroll
            for (int ks = 0; ks < 16; ++ks) {
                int k = 4 * ks + 2 * hi;
                const float* bp = W2cm + (16 * t + lo) * 64 + k;
                v2f b; b.x = bp[0]; b.y = bp[1];
                c = __builtin_amdgcn_wmma_f32_16x16x4_f32(
                        false, a[ks], false, b, (short)0, c, false, false);
            }
            const int ccol = 16 * t + lo;
            #pragma unroll
            for (int r = 0; r < 8; ++r) {
                float wv = c[r];
                const int srcb = srcs[r] * 160;
                const int dstb = dsts[r] * 384;
                if (t < 4) {                       // w0 * e0 * sh0 -> n0
                    float e0 = f[srcb + ccol];
                    atomicAdd(&acc[dstb + ccol], wv * e0 * s0[r]);
                    if (t == 0 && lo == 0) atomicAdd(&cnt[dsts[r]], 1.0f);
                } else if (t < 8) {                // w1 * e0 * sh1[m] -> n2
                    int u = ccol - 64;
                    float p = wv * f[srcb + u];
                    atomicAdd(&acc[dstb + 96 + u * 3 + 0], p * s1x[r]);
                    atomicAdd(&acc[dstb + 96 + u * 3 + 1], p * s1y[r]);
                    atomicAdd(&acc[dstb + 96 + u * 3 + 2], p * s1z[r]);
                } else if (t < 10) {               // w2 * e1 * sh0 -> n3
                    int u = ccol - 128;
                    const float* fe = f + srcb + 64 + u * 3;
                    float p = wv * s0[r];
                    atomicAdd(&acc[dstb + 96 + (64 + u) * 3 + 0], p * fe[0]);
                    atomicAdd(&acc[dstb + 96 + (64 + u) * 3 + 1], p * fe[1]);
                    atomicAdd(&acc[dstb + 96 + (64 + u) * 3 + 2], p * fe[2]);
                } else {                           // w3 * <e1,sh1>/sqrt3 -> n1
                    int u = ccol - 160;
                    const float* fe = f + srcb + 64 + u * 3;
                    float dot = fe[0] * s1x[r] + fe[1] * s1y[r] + fe[2] * s1z[r];
                    atomicAdd(&acc[dstb + 64 + u], wv * dot * inv_sqrt3);
                }
            }
        }
    }
}

// ---------------------------------------------------------------------------
// Kernel 3: segment-mean, final linears, residual combine.
// ---------------------------------------------------------------------------
__global__ __launch_bounds__(160) void final_kernel(
    const float* __restrict__ acc, const float* __restrict__ cnt,
    const float* __restrict__ sc, const float* __restrict__ na,
    const float* __restrict__ Wl2_s, const float* __restrict__ Wl2_v,
    const float* __restrict__ Wa, float* __restrict__ out) {
    __shared__ float sscal[96];
    __shared__ float svec[288];
    __shared__ float salpha;
    const int n = blockIdx.x;
    const int tid = threadIdx.x;
    float c = cnt[n];
    if (c < 1.0f) c = 1.0f;
    const float ic = 1.0f / c;
    const float* ar = acc + (size_t)n * 384;
    if (tid < 96) sscal[tid] = ar[tid] * ic;
    for (int j = tid; j < 288; j += 160) svec[j] = ar[96 + j] * ic;
    __syncthreads();
    const float a = na[n];
    const float inv96 = 0.10206207261596575449f;   // 1/sqrt(96)
    if (tid == 0) {
        float al = 0.f;
        for (int u = 0; u < 96; ++u) al += sscal[u] * Wa[u];
        salpha = al * a * inv96;
    }
    __syncthreads();
    const float alpha = salpha;
    float conv;
    int col;
    if (tid < 64) {
        float s = 0.f;
        #pragma unroll 8
        for (int u = 0; u < 96; ++u) s += sscal[u] * Wl2_s[u * 64 + tid];
        conv = s * a * inv96;
        col = tid;
    } else {
        int j = tid - 64, v = j / 3, m = j - 3 * v;
        float s = 0.f;
        #pragma unroll 8
        for (int u = 0; u < 96; ++u) s += svec[u * 3 + m] * Wl2_v[u * 32 + v];
        conv = s * a * inv96;
        col = 64 + v * 3 + m;
    }
    out[(size_t)n * 160 + col] = sc[(size_t)n * 160 + col] + alpha * conv;
}

// ---------------------------------------------------------------------------
extern "C" void kernel_launch(void* const* d_in, const int* in_sizes, int n_in,
                              void* d_out, int out_size, void* d_ws, size_t ws_size,
                              hipStream_t stream) {
    (void)in_sizes; (void)n_in; (void)out_size; (void)ws_size;
    const float* x     = (const float*)d_in[0];
    const float* na    = (const float*)d_in[1];
    const float* es    = (const float*)d_in[2];
    const float* sh    = (const float*)d_in[3];
    const int*   esrc  = (const int*)d_in[4];
    const int*   edst  = (const int*)d_in[5];
    const float* Wfc1  = (const float*)d_in[6];
    const float* Wfc2  = (const float*)d_in[7];
    const float* Wsc0  = (const float*)d_in[8];
    const float* Wsc1  = (const float*)d_in[9];
    const float* Wl1_0 = (const float*)d_in[10];
    const float* Wl1_1 = (const float*)d_in[11];
    const float* Wl2_s = (const float*)d_in[12];
    const float* Wl2_v = (const float*)d_in[13];
    const float* Wa    = (const float*)d_in[14];
    float* out = (float*)d_out;

    float* ws   = (float*)d_ws;
    float* f    = ws;                       // N*160
    float* sc   = f    + (size_t)NN * 160;  // N*160
    float* acc  = sc   + (size_t)NN * 160;  // N*384
    float* cnt  = acc  + (size_t)NN * 384;  // N
    float* W1cm = cnt  + NN;                // 1024
    float* W2cm = W1cm + 1024;              // 12288

    hipMemsetAsync(acc, 0, (size_t)(NN * 384 + NN) * sizeof(float), stream);
    wprep_kernel<<<48, 256, 0, stream>>>(Wfc1, Wfc2, W1cm, W2cm);
    node_kernel<<<NN, 160, 0, stream>>>(x, na, Wsc0, Wsc1, Wl1_0, Wl1_1, f, sc);
    edge_kernel<<<EE / 128, 128, 0, stream>>>(es, sh, esrc, edst, W1cm, W2cm, f, acc, cnt);
    final_kernel<<<NN, 160, 0, stream>>>(acc, cnt, sc, na, Wl2_s, Wl2_v, Wa, out);
}